// FaceDetetor_11879879542631
// MI455X (gfx1250) — compile-verified
//
#include <hip/hip_runtime.h>
#include <hip/hip_bf16.h>
#include <stdint.h>

// FaceDetector NMS for gfx1250 (MI455X), wave32.
// init -> scan (compact SoA or alive-bitmask; folds argmax0) -> 5x [select -> suppress+argmax(next)]
// AoS det rows staged to LDS via CDNA5 async DMA: global_load_async_to_lds_b128 + s_wait_asynccnt.

#define C17 17
#define SCORE_THR 0.5f
#define IOU_THR 0.3f
#define CLIP_MAXF 100000000.0f
#define IMG_SIZEF 128.0f
#define BLK 256

__device__ __forceinline__ unsigned long long pack_key(float score, unsigned idx) {
  // score >= 0.5 > 0 -> positive float bits order as uint. Tie-break: smaller idx wins.
  return ((unsigned long long)__float_as_uint(score) << 32) |
         (unsigned long long)(0xFFFFFFFFu - idx);
}

__device__ __forceinline__ unsigned long long warp_max_u64(unsigned long long k) {
  for (int off = 16; off > 0; off >>= 1) {
    unsigned long long o = __shfl_down(k, off, 32);
    if (o > k) k = o;
  }
  return k;
}

// Stage this block's det rows (68B AoS records) into LDS via async global->LDS DMA.
// Caller must __syncthreads() after this returns (each wave waits its own ASYNCcnt here).
__device__ __forceinline__ void stage_tile_async(const float* det, long long blockStart,
                                                 int N, float* tile) {
  int tid = threadIdx.x;
  long long remRows = (long long)N - blockStart;
  int rows = remRows < BLK ? (int)remRows : BLK;
  int bytes = rows * (C17 * 4);
  int chunks = bytes >> 4;
  uint64_t gbase = (uint64_t)(uintptr_t)det + (uint64_t)blockStart * (C17 * 4);
  unsigned ldsBase = (unsigned)(uintptr_t)tile;  // low 32 bits = LDS byte offset
  for (int c = tid; c < chunks; c += BLK) {
    uint64_t ga = gbase + ((uint64_t)c << 4);
    unsigned la = ldsBase + ((unsigned)c << 4);
    asm volatile("global_load_async_to_lds_b128 %0, %1, off"
                 :: "v"(la), "v"(ga) : "memory");
  }
  int rem = bytes & 15;  // not hit for N = 2e6 (rows % 4 == 0)
  if (rem && tid == 0) {
    for (int b = bytes - rem; b < bytes; b += 4)
      tile[b >> 2] = det[(size_t)blockStart * C17 + (b >> 2)];
  }
  asm volatile("s_wait_asynccnt 0" ::: "memory");
}

__global__ void initKernel(unsigned long long* keys, float* sel, unsigned* count) {
  int t = threadIdx.x;
  if (t < 5) keys[t] = 0ULL;
  if (t < 40) sel[t] = 0.0f;
  if (t == 0) *count = 0u;
}

// ---------------- Compacted-candidate path (needs ws >= 4096 + 24*N) ----------------

__global__ void scanCompactKernel(const float* __restrict__ det,
                                  unsigned long long* __restrict__ keys,
                                  float* __restrict__ cx1, float* __restrict__ cy1,
                                  float* __restrict__ cx2, float* __restrict__ cy2,
                                  float* __restrict__ csc, unsigned* __restrict__ corig,
                                  unsigned* __restrict__ count, int N) {
  __shared__ __align__(16) float tile[BLK * C17];
  __shared__ unsigned long long blockKey;
  long long blockStart = (long long)blockIdx.x * BLK;
  stage_tile_async(det, blockStart, N, tile);
  if (threadIdx.x == 0) blockKey = 0ULL;
  __syncthreads();

  int tid = threadIdx.x;
  unsigned lane = tid & 31u;
  long long a = blockStart + tid;
  bool ok = false;
  float score = 0.0f, x1 = 0.0f, y1 = 0.0f, x2 = 0.0f, y2 = 0.0f;
  if (a < N) {
    int r = tid * C17;
    score = tile[r + 16];
    ok = (score >= SCORE_THR);
    if (ok) {
      float cy = tile[r + 0], cx = tile[r + 1], h = tile[r + 2], w = tile[r + 3];
      x1 = fminf(fmaxf(cx - 0.5f * w, 0.0f), CLIP_MAXF);
      y1 = fminf(fmaxf(cy - 0.5f * h, 0.0f), CLIP_MAXF);
      x2 = cx + 0.5f * w;
      y2 = cy + 0.5f * h;
    }
  }
  unsigned mask = (unsigned)(__ballot(ok) & 0xFFFFFFFFull);
  int cnt = __popc(mask);
  unsigned base = 0u;
  if (lane == 0 && cnt) base = atomicAdd(count, (unsigned)cnt);
  base = __shfl(base, 0, 32);
  if (ok) {
    unsigned pos = base + (unsigned)__popc(mask & ((1u << lane) - 1u));
    cx1[pos] = x1; cy1[pos] = y1; cx2[pos] = x2; cy2[pos] = y2;
    csc[pos] = score; corig[pos] = (unsigned)a;
  }
  unsigned long long key = ok ? pack_key(score, (unsigned)a) : 0ULL;
  key = warp_max_u64(key);
  if (lane == 0 && key) atomicMax(&blockKey, key);
  __syncthreads();
  if (tid == 0 && blockKey) atomicMax(&keys[0], blockKey);
}

__global__ void suppressCompactKernel(unsigned long long* __restrict__ keys,
                                      const float* __restrict__ sel,
                                      const float* __restrict__ cx1, const float* __restrict__ cy1,
                                      const float* __restrict__ cx2, const float* __restrict__ cy2,
                                      float* __restrict__ csc, const unsigned* __restrict__ corig,
                                      const unsigned* __restrict__ count, int it) {
  const float* s = sel + it * 8;
  if (s[5] == 0.0f) return;  // no winner -> keys[it+1] stays 0
  __shared__ unsigned long long blockKey;
  if (threadIdx.x == 0) blockKey = 0ULL;
  __syncthreads();

  const float X1 = s[0], Y1 = s[1], X2 = s[2], Y2 = s[3], AREA = s[4];
  const unsigned selOrig = __float_as_uint(s[6]);
  const float NEG_INF = -__builtin_inff();

  unsigned M = *count;
  unsigned i = blockIdx.x * BLK + threadIdx.x;
  unsigned lane = threadIdx.x & 31u;
  unsigned long long key = 0ULL;
  if (i < M) {
    float score = csc[i];
    if (score > NEG_INF) {
      float x1 = cx1[i], y1 = cy1[i], x2 = cx2[i], y2 = cy2[i];
      unsigned orig = corig[i];
      float area = (x2 - x1) * (y2 - y1);
      float iw = fmaxf(fminf(x2, X2) - fmaxf(x1, X1), 0.0f);
      float ih = fmaxf(fminf(y2, Y2) - fmaxf(y1, Y1), 0.0f);
      float inter = iw * ih;
      float iou = inter / (area + AREA - inter + 1e-9f);
      if (iou > IOU_THR || orig == selOrig) {
        csc[i] = NEG_INF;
      } else {
        key = pack_key(score, orig);
      }
    }
  }
  key = warp_max_u64(key);
  if (lane == 0 && key) atomicMax(&blockKey, key);
  __syncthreads();
  if (threadIdx.x == 0 && blockKey) atomicMax(&keys[it + 1], blockKey);
}

// ---------------- Alive-bitmask fallback path (ws ~ 254 KB) ----------------

__global__ void scanBitmapKernel(const float* __restrict__ det,
                                 unsigned long long* __restrict__ keys,
                                 unsigned* __restrict__ alive, int N) {
  __shared__ unsigned long long blockKey;
  if (threadIdx.x == 0) blockKey = 0ULL;
  __syncthreads();
  long long a = (long long)blockIdx.x * BLK + threadIdx.x;
  unsigned lane = threadIdx.x & 31u;
  bool ok = false;
  float score = 0.0f;
  if (a < N) {
    score = det[(size_t)a * C17 + 16];
    ok = (score >= SCORE_THR);
  }
  unsigned m = (unsigned)(__ballot(ok) & 0xFFFFFFFFull);
  if (a < N && lane == 0) alive[(unsigned)(a >> 5)] = m;
  unsigned long long key = ok ? pack_key(score, (unsigned)a) : 0ULL;
  key = warp_max_u64(key);
  if (lane == 0 && key) atomicMax(&blockKey, key);
  __syncthreads();
  if (threadIdx.x == 0 && blockKey) atomicMax(&keys[0], blockKey);
}

__global__ void suppressBitmapKernel(const float* __restrict__ det,
                                     unsigned long long* __restrict__ keys,
                                     const float* __restrict__ sel,
                                     unsigned* __restrict__ alive, int it, int N) {
  const float* s = sel + it * 8;
  if (s[5] == 0.0f) return;
  __shared__ __align__(16) float tile[BLK * C17];
  __shared__ unsigned long long blockKey;
  const float X1 = s[0], Y1 = s[1], X2 = s[2], Y2 = s[3], AREA = s[4];
  const unsigned selOrig = __float_as_uint(s[6]);

  long long blockStart = (long long)blockIdx.x * BLK;
  stage_tile_async(det, blockStart, N, tile);
  if (threadIdx.x == 0) blockKey = 0ULL;
  __syncthreads();

  int tid = threadIdx.x;
  long long a = blockStart + tid;
  unsigned lane = tid & 31u;
  bool newAlive = false;
  float score = 0.0f;
  if (a < N) {
    unsigned word = alive[(unsigned)(a >> 5)];
    if ((word >> lane) & 1u) {
      int r = tid * C17;
      float cy = tile[r + 0], cx = tile[r + 1], h = tile[r + 2], w = tile[r + 3];
      score = tile[r + 16];
      float x1 = fminf(fmaxf(cx - 0.5f * w, 0.0f), CLIP_MAXF);
      float y1 = fminf(fmaxf(cy - 0.5f * h, 0.0f), CLIP_MAXF);
      float x2 = cx + 0.5f * w;
      float y2 = cy + 0.5f * h;
      float area = (x2 - x1) * (y2 - y1);
      float iw = fmaxf(fminf(x2, X2) - fmaxf(x1, X1), 0.0f);
      float ih = fmaxf(fminf(y2, Y2) - fmaxf(y1, Y1), 0.0f);
      float inter = iw * ih;
      float iou = inter / (area + AREA - inter + 1e-9f);
      newAlive = !(iou > IOU_THR) && ((unsigned)a != selOrig);
    }
  }
  unsigned newWord = (unsigned)(__ballot(newAlive) & 0xFFFFFFFFull);
  if (a < N && lane == 0) alive[(unsigned)(a >> 5)] = newWord;
  unsigned long long key = newAlive ? pack_key(score, (unsigned)a) : 0ULL;
  key = warp_max_u64(key);
  if (lane == 0 && key) atomicMax(&blockKey, key);
  __syncthreads();
  if (tid == 0 && blockKey) atomicMax(&keys[it + 1], blockKey);
}

// ---------------- Shared select kernel ----------------

__global__ void selectKernel(const float* __restrict__ det,
                             float* __restrict__ out,
                             const unsigned long long* __restrict__ keys,
                             float* __restrict__ sel, int it) {
  unsigned long long k = keys[it];
  bool valid = (k != 0ULL);
  unsigned orig = 0xFFFFFFFFu - (unsigned)(k & 0xFFFFFFFFull);
  int j = threadIdx.x;
  if (j < C17) {
    float v = 0.0f;
    if (valid) v = det[(size_t)orig * C17 + j];
    out[it * C17 + j] = valid ? (j < 16 ? v * IMG_SIZEF : v) : 0.0f;
  }
  if (j == 0) {
    float* s = sel + it * 8;
    if (valid) {
      size_t b = (size_t)orig * C17;
      float cy = det[b + 0], cx = det[b + 1], h = det[b + 2], w = det[b + 3];
      float x1 = fminf(fmaxf(cx - 0.5f * w, 0.0f), CLIP_MAXF);
      float y1 = fminf(fmaxf(cy - 0.5f * h, 0.0f), CLIP_MAXF);
      float x2 = cx + 0.5f * w;
      float y2 = cy + 0.5f * h;
      s[0] = x1; s[1] = y1; s[2] = x2; s[3] = y2;
      s[4] = (x2 - x1) * (y2 - y1);
      s[5] = 1.0f;
      s[6] = __uint_as_float(orig);
    } else {
      s[5] = 0.0f;
    }
  }
}

extern "C" void kernel_launch(void* const* d_in, const int* in_sizes, int n_in,
                              void* d_out, int out_size, void* d_ws, size_t ws_size,
                              hipStream_t stream) {
  const float* det = (const float*)d_in[0];
  int N = in_sizes[0] / C17;  // 2,000,000
  float* out = (float*)d_out;

  char* ws = (char*)d_ws;
  unsigned long long* keys = (unsigned long long*)ws;  // @0:    5 * 8B
  float* sel = (float*)(ws + 64);                      // @64:   5 * 8 floats
  unsigned* count = (unsigned*)(ws + 256);             // @256:  1 u32
  char* payload = ws + 4096;

  int nb = (N + BLK - 1) / BLK;
  bool useCompact = ws_size >= (size_t)4096 + (size_t)N * 24u;

  initKernel<<<1, 64, 0, stream>>>(keys, sel, count);
  if (useCompact) {
    float* cx1 = (float*)payload;
    float* cy1 = cx1 + N;
    float* cx2 = cy1 + N;
    float* cy2 = cx2 + N;
    float* csc = cy2 + N;
    unsigned* corig = (unsigned*)(csc + N);
    scanCompactKernel<<<nb, BLK, 0, stream>>>(det, keys, cx1, cy1, cx2, cy2, csc, corig, count, N);
    for (int it = 0; it < 5; ++it) {
      selectKernel<<<1, 32, 0, stream>>>(det, out, keys, sel, it);
      if (it < 4)
        suppressCompactKernel<<<nb, BLK, 0, stream>>>(keys, sel, cx1, cy1, cx2, cy2, csc, corig, count, it);
    }
  } else {
    unsigned* alive = (unsigned*)payload;  // ceil(N/32) u32
    scanBitmapKernel<<<nb, BLK, 0, stream>>>(det, keys, alive, N);
    for (int it = 0; it < 5; ++it) {
      selectKernel<<<1, 32, 0, stream>>>(det, out, keys, sel, it);
      if (it < 4)
        suppressBitmapKernel<<<nb, BLK, 0, stream>>>(det, keys, sel, alive, it, N);
    }
  }
}